// DPHGNNConv_67619965108633
// MI455X (gfx1250) — compile-verified
//
#include <hip/hip_runtime.h>
#include <math.h>

// ---------------------------------------------------------------------------
// DPHGNN hypergraph convolution for MI455X (gfx1250, wave32).
// GEMMs use V_WMMA_F32_16X16X4_F32 (exact fp32). Scatter phases rely on the
// 192MB L2: all gather sources / atomic targets are L2-resident, so the
// kernel set is organized as flat, coalesced passes with f32 L2 atomics.
// ---------------------------------------------------------------------------

typedef float v2f __attribute__((ext_vector_type(2)));
typedef float v8f __attribute__((ext_vector_type(8)));

#define N_NODES 100000
#define N_EDGES 20000
#define NNZ     1600000
#define D_IN    128
#define D_OUT   128
#define STAR    64
#define EMSG    (D_OUT + STAR)   // 192

__device__ __forceinline__ float lrelu(float x) { return x > 0.f ? x : 0.2f * x; }
__device__ __forceinline__ float eluf (float x) { return x > 0.f ? x : __expf(x) - 1.f; }

// ---------------------------------------------------------------------------
// Workspace init: zero accumulators, -inf for segment max.
// ---------------------------------------------------------------------------
__global__ void k_init(float* __restrict__ ns, float* __restrict__ ya,
                       float* __restrict__ em, float* __restrict__ den,
                       float* __restrict__ cnt)
{
    const int i = blockIdx.x * blockDim.x + threadIdx.x;
    if (i < N_NODES * D_OUT) ns[i] = 0.f;
    if (i < N_EDGES * D_OUT) ya[i] = 0.f;
    if (i < N_EDGES) { em[i] = -INFINITY; den[i] = 0.f; }
    if (i < N_NODES) cnt[i] = 0.f;
}

// ---------------------------------------------------------------------------
// Fused dual GEMM: X_init = X*Wx^T + bx ; X_feat = X*Wv^T + bv.
// One wave per 16x16 output tile (shared for both outputs -> A loaded once).
// f32 WMMA fragment layouts (ISA 7.12.2):
//   A (16x4): lane%16 = M, VGPR v holds K = v + 2*(lane/16)       -> float2 load
//   B (4x16): lane%16 = N, VGPR v holds K = v + 2*(lane/16)       -> float2 load
//   C/D     : lane%16 = N, VGPR v holds M = v + 8*(lane/16)
// B[k][n] = W[n][k] since out = X @ W^T.
// ---------------------------------------------------------------------------
__global__ __launch_bounds__(128) void k_gemm_dual(
    const float* __restrict__ X,  const float* __restrict__ Wx,
    const float* __restrict__ bx, const float* __restrict__ Wv,
    const float* __restrict__ bv, float* __restrict__ Xi, float* __restrict__ Xf)
{
    const int wave = threadIdx.x >> 5;
    const int lane = threadIdx.x & 31;
    const int tile = blockIdx.x * 4 + wave;     // 6250*8 = 50000 tiles
    const int tm = tile >> 3;                   // node tile
    const int tn = tile & 7;                    // column tile
    const int lo = lane & 15, hi = lane >> 4;

    const float* xrow = X  + (size_t)(tm * 16 + lo) * D_IN;
    const float* wxr  = Wx + (size_t)(tn * 16 + lo) * D_IN;
    const float* wvr  = Wv + (size_t)(tn * 16 + lo) * D_IN;

    v8f ci = {0.f,0.f,0.f,0.f,0.f,0.f,0.f,0.f};
    v8f cf = {0.f,0.f,0.f,0.f,0.f,0.f,0.f,0.f};
    #pragma unroll 4
    for (int k = 0; k < D_IN; k += 4) {
        const int kk = k + 2 * hi;
        v2f a  = *(const v2f*)(xrow + kk);
        v2f b0 = *(const v2f*)(wxr  + kk);
        v2f b1 = *(const v2f*)(wvr  + kk);
        ci = __builtin_amdgcn_wmma_f32_16x16x4_f32(false, a, false, b0, (short)0, ci, false, false);
        cf = __builtin_amdgcn_wmma_f32_16x16x4_f32(false, a, false, b1, (short)0, cf, false, false);
    }
    const int col = tn * 16 + lo;
    const float bxi = bx[col], bvi = bv[col];
    #pragma unroll
    for (int v = 0; v < 8; ++v) {
        const int row = tm * 16 + v + 8 * hi;
        Xi[(size_t)row * D_OUT + col] = ci[v] + bxi;
        Xf[(size_t)row * D_OUT + col] = cf[v] + bvi;
    }
}

// ---------------------------------------------------------------------------
// score[n] = dot(X_feat[n,:], a_w). One wave per node, shuffle reduction.
// ---------------------------------------------------------------------------
__global__ __launch_bounds__(256) void k_score(const float* __restrict__ Xf,
                                               const float* __restrict__ aw,
                                               float* __restrict__ score)
{
    const int wave = threadIdx.x >> 5, lane = threadIdx.x & 31;
    const int n = blockIdx.x * 8 + wave;
    if (n >= N_NODES) return;
    float s = 0.f;
    #pragma unroll
    for (int c = 0; c < 4; ++c) {
        const int d = lane + 32 * c;
        s += Xf[(size_t)n * D_OUT + d] * aw[d];
    }
    #pragma unroll
    for (int off = 16; off > 0; off >>= 1) s += __shfl_xor(s, off, 32);
    if (lane == 0) score[n] = s;
}

// ---------------------------------------------------------------------------
// Segment max of leaky-relu scores over hyperedge groups (f32 atomic max).
// ---------------------------------------------------------------------------
__global__ void k_edge_max(const int* __restrict__ V, const int* __restrict__ E,
                           const float* __restrict__ score, float* __restrict__ em)
{
    const int i = blockIdx.x * blockDim.x + threadIdx.x;
    if (i >= NNZ) return;
    atomicMax(&em[E[i]], lrelu(score[V[i]]));
}

// ---------------------------------------------------------------------------
// Softmax denominator per edge + per-node incidence counts.
// ---------------------------------------------------------------------------
__global__ void k_edge_denom(const int* __restrict__ V, const int* __restrict__ E,
                             const float* __restrict__ score, const float* __restrict__ em,
                             float* __restrict__ den, float* __restrict__ cnt)
{
    const int i = blockIdx.x * blockDim.x + threadIdx.x;
    if (i >= NNZ) return;
    const int v = V[i], e = E[i];
    atomicAdd(&den[e], __expf(lrelu(score[v]) - em[e]));
    atomicAdd(&cnt[v], 1.f);
}

// ---------------------------------------------------------------------------
// Node->edge weighted scatter. One wave per incidence: lane = column (x4),
// coalesced 128-float row reads of L2-resident X_feat, f32 atomics into the
// 10MB edge accumulator (L2-resident).
// ---------------------------------------------------------------------------
__global__ __launch_bounds__(256) void k_v2e(
    const int* __restrict__ V, const int* __restrict__ E,
    const float* __restrict__ score, const float* __restrict__ em,
    const float* __restrict__ den, const float* __restrict__ Xf,
    float* __restrict__ ya)
{
    const int wave = threadIdx.x >> 5, lane = threadIdx.x & 31;
    const int i = blockIdx.x * 8 + wave;        // NNZ/8 blocks exactly
    const int v = V[i], e = E[i];
    const float soft = __expf(lrelu(score[v]) - em[e]) / den[e];
    const float* xr = Xf + (size_t)v * D_OUT;
    float* yr = ya + (size_t)e * D_OUT;
    #pragma unroll
    for (int c = 0; c < 4; ++c) {
        const int d = lane + 32 * c;
        atomicAdd(&yr[d], xr[d] * soft);
    }
}

// ---------------------------------------------------------------------------
// Edge GEMM: Y = [elu(Yacc) || S] @ Wt^T + bt  (K = 192). The A fragment is
// built on the fly: k<128 -> elu(edge accumulator), k>=128 -> S_features.
// K step never straddles the 128 boundary (pairs are even-aligned).
// ---------------------------------------------------------------------------
__global__ __launch_bounds__(128) void k_edge_gemm(
    const float* __restrict__ ya, const float* __restrict__ S,
    const float* __restrict__ Wt, const float* __restrict__ bt,
    float* __restrict__ Y)
{
    const int wave = threadIdx.x >> 5, lane = threadIdx.x & 31;
    const int tile = blockIdx.x * 4 + wave;     // 1250*8 = 10000 tiles
    const int tm = tile >> 3, tn = tile & 7;
    const int lo = lane & 15, hi = lane >> 4;
    const int m = tm * 16 + lo;

    const float* yrow = ya + (size_t)m * D_OUT;
    const float* srow = S  + (size_t)m * STAR;
    const float* wrow = Wt + (size_t)(tn * 16 + lo) * EMSG;

    v8f c = {0.f,0.f,0.f,0.f,0.f,0.f,0.f,0.f};
    #pragma unroll 4
    for (int k = 0; k < EMSG; k += 4) {
        const int kk = k + 2 * hi;
        v2f a;
        if (kk < D_OUT) {
            a[0] = eluf(yrow[kk]);
            a[1] = eluf(yrow[kk + 1]);
        } else {
            a = *(const v2f*)(srow + (kk - D_OUT));
        }
        v2f b = *(const v2f*)(wrow + kk);
        c = __builtin_amdgcn_wmma_f32_16x16x4_f32(false, a, false, b, (short)0, c, false, false);
    }
    const int col = tn * 16 + lo;
    const float bb = bt[col];
    #pragma unroll
    for (int v = 0; v < 8; ++v)
        Y[(size_t)(tm * 16 + v + 8 * hi) * D_OUT + col] = c[v] + bb;
}

// ---------------------------------------------------------------------------
// Edge->node scatter: gather Y rows (10MB, L2-resident), atomic add into
// node accumulator. Same wave-per-incidence structure.
// ---------------------------------------------------------------------------
__global__ __launch_bounds__(256) void k_e2v(
    const int* __restrict__ V, const int* __restrict__ E,
    const float* __restrict__ Y, float* __restrict__ ns)
{
    const int wave = threadIdx.x >> 5, lane = threadIdx.x & 31;
    const int i = blockIdx.x * 8 + wave;
    const int v = V[i], e = E[i];
    const float* yr = Y + (size_t)e * D_OUT;
    float* nr = ns + (size_t)v * D_OUT;
    #pragma unroll
    for (int c = 0; c < 4; ++c) {
        const int d = lane + 32 * c;
        atomicAdd(&nr[d], yr[d]);
    }
}

// ---------------------------------------------------------------------------
// Finalize: mean over incidences, elu, residual add with X_init.
// ---------------------------------------------------------------------------
__global__ void k_final(const float* __restrict__ ns, const float* __restrict__ cnt,
                        const float* __restrict__ Xi, float* __restrict__ out)
{
    const int i = blockIdx.x * blockDim.x + threadIdx.x;
    if (i >= N_NODES * D_OUT) return;
    const int n = i >> 7;
    const float x = ns[i] / fmaxf(cnt[n], 1.f);
    out[i] = eluf(x) + Xi[i];
}

// ---------------------------------------------------------------------------
extern "C" void kernel_launch(void* const* d_in, const int* in_sizes, int n_in,
                              void* d_out, int out_size, void* d_ws, size_t ws_size,
                              hipStream_t stream)
{
    (void)in_sizes; (void)n_in; (void)out_size; (void)ws_size;
    const float* X  = (const float*)d_in[0];
    const int*   V  = (const int*)  d_in[1];
    const int*   E  = (const int*)  d_in[2];
    const float* S  = (const float*)d_in[3];
    const float* Wx = (const float*)d_in[4];
    const float* bx = (const float*)d_in[5];
    const float* Wv = (const float*)d_in[6];
    const float* bv = (const float*)d_in[7];
    const float* aw = (const float*)d_in[8];
    const float* Wt = (const float*)d_in[9];
    const float* bt = (const float*)d_in[10];
    float* out = (float*)d_out;

    char* ws = (char*)d_ws;
    size_t off = 0;
    auto alloc = [&](size_t bytes) -> void* {
        void* p = ws + off;
        off += (bytes + 255) & ~(size_t)255;
        return p;
    };
    float* Xi    = (float*)alloc((size_t)N_NODES * D_OUT * 4);  // X_init
    float* Xf    = (float*)alloc((size_t)N_NODES * D_OUT * 4);  // X_feat
    float* score = (float*)alloc((size_t)N_NODES * 4);
    float* em    = (float*)alloc((size_t)N_EDGES * 4);          // segment max
    float* den   = (float*)alloc((size_t)N_EDGES * 4);          // softmax denom
    float* ya    = (float*)alloc((size_t)N_EDGES * D_OUT * 4);  // edge accumulator
    float* Y     = (float*)alloc((size_t)N_EDGES * D_OUT * 4);  // transformed edges
    float* ns    = (float*)alloc((size_t)N_NODES * D_OUT * 4);  // node accumulator
    float* cnt   = (float*)alloc((size_t)N_NODES * 4);

    k_init      <<<(N_NODES * D_OUT + 255) / 256, 256, 0, stream>>>(ns, ya, em, den, cnt);
    k_gemm_dual <<<12500, 128, 0, stream>>>(X, Wx, bx, Wv, bv, Xi, Xf);
    k_score     <<<12500, 256, 0, stream>>>(Xf, aw, score);
    k_edge_max  <<<NNZ / 256, 256, 0, stream>>>(V, E, score, em);
    k_edge_denom<<<NNZ / 256, 256, 0, stream>>>(V, E, score, em, den, cnt);
    k_v2e       <<<NNZ / 8, 256, 0, stream>>>(V, E, score, em, den, Xf, ya);
    k_edge_gemm <<<2500, 128, 0, stream>>>(ya, S, Wt, bt, Y);
    k_e2v       <<<NNZ / 8, 256, 0, stream>>>(V, E, Y, ns);
    k_final     <<<(N_NODES * D_OUT + 255) / 256, 256, 0, stream>>>(ns, cnt, Xi, out);
}